// GAT_21191368639149
// MI455X (gfx1250) — compile-verified
//
#include <hip/hip_runtime.h>
#include <math.h>

#define NN 50000
#define NE 800000
#define FIN 64
#define HEADS 4
#define DHEAD 32
#define HIDC 128          // H*D = HID = 128 channels
#define NGRAPH 1024
#define NEG_SLOPE 0.2f

typedef __attribute__((ext_vector_type(2))) float v2f;
typedef __attribute__((ext_vector_type(8))) float v8f;

// ---------------- utility ----------------
__global__ void fill_kernel(float* __restrict__ p, float v, int n) {
  int i = blockIdx.x * blockDim.x + threadIdx.x;
  if (i < n) p[i] = v;
}

__device__ __forceinline__ void atomicMaxFloat(float* addr, float v) {
  // valid for mixed-sign IEEE754 when initialized to -inf
  if (v >= 0.f) atomicMax((int*)addr, __float_as_int(v));
  else          atomicMin((unsigned int*)addr, __float_as_uint(v));
}

// ---------------- dual GEMM: xl = X@Wl + bl, xr = X@Wr + br ----------------
// One wave32 per 16x16 output tile, f32 WMMA, K-loop step 4.
__global__ void gemm_dual_wmma(const float* __restrict__ X, int K,
                               const float* __restrict__ Wl, const float* __restrict__ bl,
                               const float* __restrict__ Wr, const float* __restrict__ br,
                               float* __restrict__ xl, float* __restrict__ xr) {
  const int wave = (blockIdx.x * blockDim.x + threadIdx.x) >> 5;
  const int lane = threadIdx.x & 31;
  const int mtile = wave >> 3;          // 128/16 = 8 column tiles
  const int ntile = wave & 7;
  const int m0 = mtile * 16;
  const int n0 = ntile * 16;
  const int rc   = lane & 15;           // A-row / B-col / C-col
  const int half = lane >> 4;           // K half select

  v8f accL = {0.f,0.f,0.f,0.f,0.f,0.f,0.f,0.f};
  v8f accR = {0.f,0.f,0.f,0.f,0.f,0.f,0.f,0.f};

  const float* xrow = X + (size_t)(m0 + rc) * K;
  for (int k0 = 0; k0 < K; k0 += 4) {
    const int ka = k0 + half * 2;
    v2f a;
    a.x = xrow[ka];
    a.y = xrow[ka + 1];
    v2f bL, bR;
    bL.x = Wl[(size_t)ka * HIDC + n0 + rc];
    bL.y = Wl[(size_t)(ka + 1) * HIDC + n0 + rc];
    bR.x = Wr[(size_t)ka * HIDC + n0 + rc];
    bR.y = Wr[(size_t)(ka + 1) * HIDC + n0 + rc];
    accL = __builtin_amdgcn_wmma_f32_16x16x4_f32(false, a, false, bL, (short)0, accL, false, false);
    accR = __builtin_amdgcn_wmma_f32_16x16x4_f32(false, a, false, bR, (short)0, accR, false, false);
  }
  const float biasL = bl[n0 + rc];
  const float biasR = br[n0 + rc];
#pragma unroll
  for (int i = 0; i < 8; ++i) {
    const int m = m0 + i + 8 * half;
    xl[(size_t)m * HIDC + n0 + rc] = accL[i] + biasL;
    xr[(size_t)m * HIDC + n0 + rc] = accR[i] + biasR;
  }
}

// ---------------- edge pass 1: scores + segment max ----------------
// One wave32 per edge; lane = d within each head; 4 heads handled per lane.
__global__ void edge_score_kernel(const float* __restrict__ xl, const float* __restrict__ xr,
                                  const float* __restrict__ att, const int* __restrict__ ei,
                                  float* __restrict__ score, float* __restrict__ smax) {
  const int e = (blockIdx.x * blockDim.x + threadIdx.x) >> 5;
  const int lane = threadIdx.x & 31;
  if (e >= NE) return;
  const int src = ei[e];
  const int dst = ei[NE + e];
  const float* pl = xl + (size_t)src * HIDC;
  const float* pr = xr + (size_t)dst * HIDC;

  float m, t, s0, s1, s2, s3;
  m = pl[lane]      + pr[lane];      t = (m >= 0.f) ? m : NEG_SLOPE * m; s0 = t * att[lane];
  m = pl[32 + lane] + pr[32 + lane]; t = (m >= 0.f) ? m : NEG_SLOPE * m; s1 = t * att[32 + lane];
  m = pl[64 + lane] + pr[64 + lane]; t = (m >= 0.f) ? m : NEG_SLOPE * m; s2 = t * att[64 + lane];
  m = pl[96 + lane] + pr[96 + lane]; t = (m >= 0.f) ? m : NEG_SLOPE * m; s3 = t * att[96 + lane];

#pragma unroll
  for (int off = 16; off; off >>= 1) {
    s0 += __shfl_xor(s0, off, 32);
    s1 += __shfl_xor(s1, off, 32);
    s2 += __shfl_xor(s2, off, 32);
    s3 += __shfl_xor(s3, off, 32);
  }
  if (lane == 0) {
    float* sp = score + (size_t)e * 4;
    sp[0] = s0; sp[1] = s1; sp[2] = s2; sp[3] = s3;
    float* mp = smax + (size_t)dst * 4;
    atomicMaxFloat(&mp[0], s0);
    atomicMaxFloat(&mp[1], s1);
    atomicMaxFloat(&mp[2], s2);
    atomicMaxFloat(&mp[3], s3);
  }
}

// ---------------- edge pass 2: unnormalized weighted scatter-add ----------------
__global__ void edge_agg_kernel(const float* __restrict__ xl, const float* __restrict__ score,
                                const float* __restrict__ smax, const int* __restrict__ ei,
                                float* __restrict__ denom, float* __restrict__ agg) {
  const int e = (blockIdx.x * blockDim.x + threadIdx.x) >> 5;
  const int lane = threadIdx.x & 31;
  if (e >= NE) return;
  const int src = ei[e];
  const int dst = ei[NE + e];
  const float* sp = score + (size_t)e * 4;
  const float* mp = smax + (size_t)dst * 4;
  const float e0 = __expf(sp[0] - mp[0]);
  const float e1 = __expf(sp[1] - mp[1]);
  const float e2 = __expf(sp[2] - mp[2]);
  const float e3 = __expf(sp[3] - mp[3]);
  if (lane == 0) {
    float* dp = denom + (size_t)dst * 4;
    atomicAdd(&dp[0], e0);
    atomicAdd(&dp[1], e1);
    atomicAdd(&dp[2], e2);
    atomicAdd(&dp[3], e3);
  }
  const float* pl = xl + (size_t)src * HIDC;
  float* pa = agg + (size_t)dst * HIDC;
  atomicAdd(&pa[lane],      e0 * pl[lane]);
  atomicAdd(&pa[32 + lane], e1 * pl[32 + lane]);
  atomicAdd(&pa[64 + lane], e2 * pl[64 + lane]);
  atomicAdd(&pa[96 + lane], e3 * pl[96 + lane]);
}

// ---------------- per-node finalize: normalize + bias + ELU (in place) ----------------
__global__ void finalize_kernel(float* __restrict__ agg, const float* __restrict__ denom,
                                const float* __restrict__ b) {
  const int i = blockIdx.x * blockDim.x + threadIdx.x;
  if (i >= NN * HIDC) return;
  const int node = i >> 7;
  const int c = i & 127;
  const int h = c >> 5;
  const float d = denom[node * 4 + h];
  float v = (d > 0.f) ? (agg[i] / d) : 0.f;   // no in-edges -> zero aggregation
  v += b[c];
  agg[i] = (v > 0.f) ? v : (__expf(v) - 1.f); // ELU
}

// ---------------- global mean pool + head ----------------
__global__ void pool_accum_kernel(const float* __restrict__ h, const int* __restrict__ batch,
                                  float* __restrict__ psum, float* __restrict__ pcnt) {
  const int i = blockIdx.x * blockDim.x + threadIdx.x;
  if (i >= NN * HIDC) return;
  const int node = i >> 7;
  const int c = i & 127;
  const int g = batch[node];
  atomicAdd(&psum[(size_t)g * HIDC + c], h[i]);
  if (c == 0) atomicAdd(&pcnt[g], 1.f);
}

__global__ void pool_out_kernel(const float* __restrict__ psum, const float* __restrict__ pcnt,
                                const float* __restrict__ Wh, const float* __restrict__ bh,
                                float* __restrict__ out) {
  const int g = (blockIdx.x * blockDim.x + threadIdx.x) >> 5;
  const int lane = threadIdx.x & 31;
  if (g >= NGRAPH) return;
  const float inv = 1.f / fmaxf(pcnt[g], 1.f);
  const float* ps = psum + (size_t)g * HIDC;
  float acc = 0.f;
#pragma unroll
  for (int j = 0; j < 4; ++j) {
    const int c = 32 * j + lane;
    acc += (ps[c] * inv) * Wh[c];
  }
#pragma unroll
  for (int off = 16; off; off >>= 1) acc += __shfl_xor(acc, off, 32);
  if (lane == 0) out[g] = acc + bh[0];
}

// ---------------- host ----------------
extern "C" void kernel_launch(void* const* d_in, const int* in_sizes, int n_in,
                              void* d_out, int out_size, void* d_ws, size_t ws_size,
                              hipStream_t stream) {
  const float* x     = (const float*)d_in[0];
  const int*   ei    = (const int*)d_in[1];     // [2, E]
  const int*   batch = (const int*)d_in[2];
  const float* Wl[3] = {(const float*)d_in[3],  (const float*)d_in[9],  (const float*)d_in[15]};
  const float* bl[3] = {(const float*)d_in[4],  (const float*)d_in[10], (const float*)d_in[16]};
  const float* Wr[3] = {(const float*)d_in[5],  (const float*)d_in[11], (const float*)d_in[17]};
  const float* br[3] = {(const float*)d_in[6],  (const float*)d_in[12], (const float*)d_in[18]};
  const float* at[3] = {(const float*)d_in[7],  (const float*)d_in[13], (const float*)d_in[19]};
  const float* bb[3] = {(const float*)d_in[8],  (const float*)d_in[14], (const float*)d_in[20]};
  const float* Wh    = (const float*)d_in[21];
  const float* bh    = (const float*)d_in[22];
  float* out = (float*)d_out;

  float* ws = (float*)d_ws;
  size_t off = 0;
  float* xl    = ws + off; off += (size_t)NN * HIDC;
  float* xr    = ws + off; off += (size_t)NN * HIDC;
  float* score = ws + off; off += (size_t)NE * HEADS;
  float* smax  = ws + off; off += (size_t)NN * HEADS;
  float* denom = ws + off; off += (size_t)NN * HEADS;
  float* bufA  = ws + off; off += (size_t)NN * HIDC;
  float* bufB  = ws + off; off += (size_t)NN * HIDC;
  float* psum  = ws + off; off += (size_t)NGRAPH * HIDC;
  float* pcnt  = ws + off; off += (size_t)NGRAPH;

  const int gemm_blocks = (NN / 16) * 8 / 4;        // 4 waves (tiles) per 128-thread block
  const int edge_blocks = (NE + 7) / 8;             // 8 waves per 256-thread block
  const int nc  = NN * HIDC;
  const int nh  = NN * HEADS;
  const float ninf = -__builtin_inff();

  const float* hin = x;
  int K = FIN;
  float* houts[3] = {bufA, bufB, bufA};
  for (int l = 0; l < 3; ++l) {
    float* hout = houts[l];
    gemm_dual_wmma<<<gemm_blocks, 128, 0, stream>>>(hin, K, Wl[l], bl[l], Wr[l], br[l], xl, xr);
    fill_kernel<<<(nh + 255) / 256, 256, 0, stream>>>(smax, ninf, nh);
    fill_kernel<<<(nh + 255) / 256, 256, 0, stream>>>(denom, 0.f, nh);
    fill_kernel<<<(nc + 255) / 256, 256, 0, stream>>>(hout, 0.f, nc);
    edge_score_kernel<<<edge_blocks, 256, 0, stream>>>(xl, xr, at[l], ei, score, smax);
    edge_agg_kernel<<<edge_blocks, 256, 0, stream>>>(xl, score, smax, ei, denom, hout);
    finalize_kernel<<<(nc + 255) / 256, 256, 0, stream>>>(hout, denom, bb[l]);
    hin = hout;
    K = HIDC;
  }

  const int ng = NGRAPH * HIDC;
  fill_kernel<<<(ng + 255) / 256, 256, 0, stream>>>(psum, 0.f, ng);
  fill_kernel<<<(NGRAPH + 255) / 256, 256, 0, stream>>>(pcnt, 0.f, NGRAPH);
  pool_accum_kernel<<<(nc + 255) / 256, 256, 0, stream>>>(bufA, batch, psum, pcnt);
  pool_out_kernel<<<(NGRAPH * 32 + 255) / 256, 256, 0, stream>>>(psum, pcnt, Wh, bh, out);
}